// PostProcessRefineMulti_48816598286446
// MI455X (gfx1250) — compile-verified
//
#include <hip/hip_runtime.h>
#include <hip/hip_bf16.h>
#include <math.h>

// Problem shape (fixed by the reference)
#define BB 16
#define QQ 3000
#define CC 1203

#define CTILE 256         // threads per block
#define CTILES 5          // ceil(1203/256)
#define CPAD 1280         // padded C for scratch rows
#define QCHUNK 120
#define QCH 25            // 25 * 120 = 3000
#define PFD 16            // prefetch distance in q-rows

#if defined(__gfx1250__) && __has_builtin(__builtin_amdgcn_global_load_async_to_lds_b32) && __has_builtin(__builtin_amdgcn_s_wait_asynccnt)
#define USE_ASYNC_LDS 1
#else
#define USE_ASYNC_LDS 0
#endif

typedef __attribute__((address_space(1))) int* gbl_i32_ptr;
typedef __attribute__((address_space(3))) int* lds_i32_ptr;

// ---------------------------------------------------------------------------
// Kernel A: partial max over Q-chunks.  partial[(b*QCH+qc)*CPAD + c]
// ---------------------------------------------------------------------------
__global__ void ppr_partial_max(const float* __restrict__ logits,
                                float* __restrict__ partial) {
  const int ct = blockIdx.x;   // C tile
  const int qc = blockIdx.y;   // Q chunk
  const int b  = blockIdx.z;   // batch
  const int c  = ct * CTILE + (int)threadIdx.x;
  float m = -INFINITY;
  if (c < CC) {
    const float* p = logits + ((size_t)b * QQ + (size_t)qc * QCHUNK) * CC + c;
    #pragma unroll 4
    for (int i = 0; i < QCHUNK; ++i) {
      if (i + PFD < QCHUNK)
        __builtin_prefetch(p + (size_t)PFD * CC, 0, 0);  // global_prefetch_b8
      m = fmaxf(m, __builtin_nontemporal_load(p));        // stream-once read
      p += CC;
    }
  }
  partial[((size_t)b * QCH + qc) * CPAD + (size_t)ct * CTILE + threadIdx.x] = m;
}

// ---------------------------------------------------------------------------
// Kernel B: reduce partials, thr[b*C+c] = 0.5 * sigmoid(max_q logit)
// ---------------------------------------------------------------------------
__global__ void ppr_finalize_thr(const float* __restrict__ partial,
                                 float* __restrict__ thr) {
  const int ct = (int)blockIdx.x % CTILES;
  const int b  = (int)blockIdx.x / CTILES;
  const int c  = ct * CTILE + (int)threadIdx.x;
  if (c >= CC) return;
  float m = -INFINITY;
  #pragma unroll 5
  for (int qc = 0; qc < QCH; ++qc)
    m = fmaxf(m, partial[((size_t)b * QCH + qc) * CPAD + c]);
  const float top = 1.0f / (1.0f + __expf(-m));  // sigmoid monotone => exact
  thr[(size_t)b * CC + c] = 0.5f * top;
}

// ---------------------------------------------------------------------------
// Kernel C: one (b,q) row per block.  Async-stage logits row to LDS, compute
// sigmoid/keep, NT-store scores+keep, any-over-C -> scaled boxes.
// ---------------------------------------------------------------------------
__global__ void ppr_main(const float* __restrict__ logits,
                         const float* __restrict__ boxes_in,
                         const int* __restrict__ class_present,
                         const float* __restrict__ thr,
                         float* __restrict__ scores,
                         float* __restrict__ keepf,
                         float* __restrict__ boxes_out) {
  __shared__ float row[CPAD];
  __shared__ int any_keep;
  const int r   = (int)blockIdx.x;   // 0 .. B*Q-1
  const int b   = r / QQ;
  const int tid = (int)threadIdx.x;
  const size_t base = (size_t)r * CC;

  if (tid == 0) any_keep = 0;

#if USE_ASYNC_LDS
  #pragma unroll
  for (int k = 0; k < CTILES; ++k) {
    const int c = tid + k * CTILE;
    if (c < CC) {
      float* gsrc = const_cast<float*>(logits + base + c);
      __builtin_amdgcn_global_load_async_to_lds_b32(
          (gbl_i32_ptr)gsrc,
          (lds_i32_ptr)(&row[c]),
          /*offset=*/0, /*cpol=*/0);
    }
  }
  __builtin_amdgcn_s_wait_asynccnt(0);   // each lane reads only what it loaded
  __asm__ volatile("" ::: "memory");
#else
  #pragma unroll
  for (int k = 0; k < CTILES; ++k) {
    const int c = tid + k * CTILE;
    if (c < CC) row[c] = __builtin_nontemporal_load(logits + base + c);
  }
#endif
  __syncthreads();  // publishes any_keep = 0 (row entries are lane-private)

  int acc = 0;
  #pragma unroll
  for (int k = 0; k < CTILES; ++k) {
    const int c = tid + k * CTILE;
    if (c < CC) {
      const float x = row[c];
      const float p = 1.0f / (1.0f + __expf(-x));
      const int keep = (p >= thr[(size_t)b * CC + c]) &
                       (class_present[(size_t)b * CC + c] != 0);
      acc |= keep;
      __builtin_nontemporal_store(keep ? p : 0.0f, &scores[base + c]);
      __builtin_nontemporal_store(keep ? 1.0f : 0.0f, &keepf[base + c]);
    }
  }
  if (acc) any_keep = 1;   // benign race: all writers store 1
  __syncthreads();
  if (tid < 4) {
    const float mul = any_keep ? 1.0f : 0.0f;
    boxes_out[(size_t)r * 4 + tid] = boxes_in[(size_t)r * 4 + tid] * mul;
  }
}

// ---------------------------------------------------------------------------
// Host entry
// ---------------------------------------------------------------------------
extern "C" void kernel_launch(void* const* d_in, const int* in_sizes, int n_in,
                              void* d_out, int out_size, void* d_ws, size_t ws_size,
                              hipStream_t stream) {
  (void)in_sizes; (void)n_in; (void)out_size; (void)ws_size;

  const float* logits = (const float*)d_in[0];      // [B,Q,C] f32
  const float* boxes  = (const float*)d_in[1];      // [B,Q,4] f32
  /* d_in[2] = target_sizes: unused by the reference math */
  const int*   cls    = (const int*)d_in[3];        // [B,C] bool-as-int

  float* out    = (float*)d_out;
  float* scores = out;                                   // [B,Q,C]
  float* keepf  = out + (size_t)BB * QQ * CC;            // [B,Q,C]
  float* boxout = out + (size_t)2 * BB * QQ * CC;        // [B,Q,4]

  float* partial = (float*)d_ws;                         // B*QCH*CPAD floats (~2.0 MB)
  float* thr     = partial + (size_t)BB * QCH * CPAD;    // B*C floats (~77 KB)

  dim3 gA(CTILES, QCH, BB);
  ppr_partial_max<<<gA, CTILE, 0, stream>>>(logits, partial);
  ppr_finalize_thr<<<BB * CTILES, CTILE, 0, stream>>>(partial, thr);
  ppr_main<<<BB * QQ, CTILE, 0, stream>>>(logits, boxes, cls, thr,
                                          scores, keepf, boxout);
}